// Router_18476949307969
// MI455X (gfx1250) — compile-verified
//
#include <hip/hip_runtime.h>
#include <hip/hip_bf16.h>

typedef float v2f __attribute__((ext_vector_type(2)));
typedef float v8f __attribute__((ext_vector_type(8)));

#define D_MODEL        2048
#define N_EXP          64
#define TOP_K          8
#define INV_T          10.0f
#define TOKENS_PER_WAVE 32
#define WAVES_PER_BLOCK 4
#define TPB            (WAVES_PER_BLOCK * 32)
#define LDS_STRIDE     65   // 64 + 1 pad -> bank (row + col) % 64, conflict-free row reads

__global__ __launch_bounds__(TPB) void router_wmma_kernel(
    const float* __restrict__ x,      // [32768, 2048]
    const float* __restrict__ W,      // [64, 2048]
    const float* __restrict__ bvec,   // [64]
    float* __restrict__ out_logits,   // [32768, 64]
    float* __restrict__ out_probs,    // [32768, 8]
    float* __restrict__ out_idx)      // [32768, 8] (indices as float)
{
    __shared__ float lds[WAVES_PER_BLOCK][TOKENS_PER_WAVE][LDS_STRIDE];

    const int lane = threadIdx.x & 31;
    const int wave = threadIdx.x >> 5;
    const int lm   = lane & 15;      // row-within-tile / col-within-tile
    const int half = lane >> 4;      // K-half selector for fp32 WMMA A/B layout
    const int tokenBase = blockIdx.x * (WAVES_PER_BLOCK * TOKENS_PER_WAVE)
                        + wave * TOKENS_PER_WAVE;

    // ---- GEMM: 32 tokens x 64 experts per wave, fp32 WMMA 16x16x4 ----
    v8f acc[2][4];
#pragma unroll
    for (int mt = 0; mt < 2; ++mt)
#pragma unroll
        for (int nt = 0; nt < 4; ++nt)
            acc[mt][nt] = (v8f){0.f, 0.f, 0.f, 0.f, 0.f, 0.f, 0.f, 0.f};

    // A layout (16x4 fp32): lane (half*16 + lm) holds row M=lm, K = k0 + 2*half + {0,1}
    const float* aPtr0 = x + (size_t)(tokenBase + lm) * D_MODEL + 2 * half;
    const float* aPtr1 = aPtr0 + (size_t)16 * D_MODEL;
    // B layout (4x16 fp32): lane holds col N=lm, K = k0 + 2*half + {0,1}; B[k][n] = W[n][k]
    const float* bPtr0 = W + (size_t)(lm)      * D_MODEL + 2 * half;
    const float* bPtr1 = W + (size_t)(16 + lm) * D_MODEL + 2 * half;
    const float* bPtr2 = W + (size_t)(32 + lm) * D_MODEL + 2 * half;
    const float* bPtr3 = W + (size_t)(48 + lm) * D_MODEL + 2 * half;

#pragma unroll 4
    for (int k = 0; k < D_MODEL; k += 4) {
        v2f a0 = *(const v2f*)aPtr0;  aPtr0 += 4;
        v2f a1 = *(const v2f*)aPtr1;  aPtr1 += 4;
        v2f b0 = *(const v2f*)bPtr0;  bPtr0 += 4;
        v2f b1 = *(const v2f*)bPtr1;  bPtr1 += 4;
        v2f b2 = *(const v2f*)bPtr2;  bPtr2 += 4;
        v2f b3 = *(const v2f*)bPtr3;  bPtr3 += 4;

        acc[0][0] = __builtin_amdgcn_wmma_f32_16x16x4_f32(false, a0, false, b0, (short)0, acc[0][0], false, false);
        acc[1][0] = __builtin_amdgcn_wmma_f32_16x16x4_f32(false, a1, false, b0, (short)0, acc[1][0], false, false);
        acc[0][1] = __builtin_amdgcn_wmma_f32_16x16x4_f32(false, a0, false, b1, (short)0, acc[0][1], false, false);
        acc[1][1] = __builtin_amdgcn_wmma_f32_16x16x4_f32(false, a1, false, b1, (short)0, acc[1][1], false, false);
        acc[0][2] = __builtin_amdgcn_wmma_f32_16x16x4_f32(false, a0, false, b2, (short)0, acc[0][2], false, false);
        acc[1][2] = __builtin_amdgcn_wmma_f32_16x16x4_f32(false, a1, false, b2, (short)0, acc[1][2], false, false);
        acc[0][3] = __builtin_amdgcn_wmma_f32_16x16x4_f32(false, a0, false, b3, (short)0, acc[0][3], false, false);
        acc[1][3] = __builtin_amdgcn_wmma_f32_16x16x4_f32(false, a1, false, b3, (short)0, acc[1][3], false, false);
    }

    // ---- Epilogue: bias + temperature, spill logit tile to LDS ----
    float bias[4];
#pragma unroll
    for (int nt = 0; nt < 4; ++nt) bias[nt] = bvec[16 * nt + lm];

    // C/D layout: VGPR r -> M = r + 8*half (rows), N = lm (cols)
#pragma unroll
    for (int mt = 0; mt < 2; ++mt) {
#pragma unroll
        for (int nt = 0; nt < 4; ++nt) {
#pragma unroll
            for (int r = 0; r < 8; ++r) {
                const int row = 16 * mt + r + 8 * half;
                const int col = 16 * nt + lm;
                lds[wave][row][col] = (acc[mt][nt][r] + bias[nt]) * INV_T;
            }
        }
    }
    __syncthreads();

    // ---- Coalesced logits store from LDS ----
    {
        float* dst = out_logits + (size_t)tokenBase * N_EXP;
#pragma unroll 4
        for (int j = 0; j < 64; ++j) {
            const int flat = j * 32 + lane;
            dst[flat] = lds[wave][flat >> 6][flat & 63];
        }
    }

    // ---- Per-token top-8 + renormalized softmax (one token per lane) ----
    {
        const float* myrow = lds[wave][lane];
        float tv[TOP_K];
        int   ti[TOP_K];
#pragma unroll
        for (int k = 0; k < TOP_K; ++k) { tv[k] = -__builtin_inff(); ti[k] = 0; }

        for (int i = 0; i < N_EXP; ++i) {
            const float v = myrow[i];
            if (v > tv[TOP_K - 1]) {   // strict > keeps lowest-index-first tie order
                int j = TOP_K - 1;
                while (j > 0 && v > tv[j - 1]) {
                    tv[j] = tv[j - 1];
                    ti[j] = ti[j - 1];
                    --j;
                }
                tv[j] = v;
                ti[j] = i;
            }
        }

        // Normalized top-k probs: full-softmax denominator cancels.
        const float mx = tv[0];
        float e[TOP_K];
        float s = 0.f;
#pragma unroll
        for (int k = 0; k < TOP_K; ++k) { e[k] = __expf(tv[k] - mx); s += e[k]; }
        const float inv = 1.0f / s;

        const int token = tokenBase + lane;
#pragma unroll
        for (int k = 0; k < TOP_K; ++k) {
            out_probs[(size_t)token * TOP_K + k] = e[k] * inv;
            out_idx  [(size_t)token * TOP_K + k] = (float)ti[k];
        }
    }
}

extern "C" void kernel_launch(void* const* d_in, const int* in_sizes, int n_in,
                              void* d_out, int out_size, void* d_ws, size_t ws_size,
                              hipStream_t stream) {
    const float* x = (const float*)d_in[0];   // [32768, 2048]
    const float* W = (const float*)d_in[1];   // [64, 2048]
    const float* b = (const float*)d_in[2];   // [64]

    const size_t T = 32768;
    float* out_logits = (float*)d_out;                       // T*64
    float* out_probs  = out_logits + T * (size_t)N_EXP;      // T*8
    float* out_idx    = out_probs  + T * (size_t)TOP_K;      // T*8

    const int tokens_per_block = WAVES_PER_BLOCK * TOKENS_PER_WAVE;  // 128
    dim3 grid((unsigned)(T / tokens_per_block));                     // 256
    dim3 block(TPB);                                                 // 128
    router_wmma_kernel<<<grid, block, 0, stream>>>(x, W, b, out_logits, out_probs, out_idx);
}